// GATModel_21242908246160
// MI455X (gfx1250) — compile-verified
//
#include <hip/hip_runtime.h>

#define IN_F      128
#define HC        128   // HEADS * OUT_F
#define HEADS     4
#define OUT_F     32
#define NEG_SLOPE 0.2f

typedef float v2f __attribute__((ext_vector_type(2)));
typedef float v4f __attribute__((ext_vector_type(4)));
typedef float v8f __attribute__((ext_vector_type(8)));

// ---------- monotonic f32 <-> u32 key for atomic max ----------
__device__ __forceinline__ unsigned fkey(float f) {
    unsigned b = __float_as_uint(f);
    return (b & 0x80000000u) ? ~b : (b | 0x80000000u);
}
__device__ __forceinline__ float funkey(unsigned k) {
    unsigned b = (k & 0x80000000u) ? (k & 0x7fffffffu) : ~k;
    return __uint_as_float(b);
}
__device__ __forceinline__ float leaky(float x) {
    return x > 0.0f ? x : NEG_SLOPE * x;
}

// ---------- kernel 0: init out = bias, mkey = 0 (== -NaN key), denom = 0 ----------
__global__ __launch_bounds__(256) void gat_init(float* __restrict__ out,
                                                const float* __restrict__ bias,
                                                unsigned* __restrict__ mkey,
                                                float* __restrict__ denom,
                                                int total, int nm) {
    int i = blockIdx.x * 256 + threadIdx.x;
    if (i < total) out[i] = bias[i & (HC - 1)];
    if (i < nm) { mkey[i] = 0u; denom[i] = 0.0f; }
}

// ---------- kernel 1: h = x @ W via V_WMMA_F32_16X16X4_F32 ----------
// block = 128 threads (4 waves); block computes rows [16*blockIdx, +16), all 128 cols.
// Wave w computes column tiles w and w+4. W and the x tile staged in padded LDS
// (strides chosen so both 16-lane halves of a wave hit disjoint bank sets).
#define WSTR 136
#define XSTR 132
#define GEMM_LDS_BYTES ((128 * WSTR + 16 * XSTR) * 4)

__global__ __launch_bounds__(128) void gat_gemm_wmma(const float* __restrict__ x,
                                                     const float* __restrict__ W,
                                                     float* __restrict__ h) {
    extern __shared__ float smem[];
    float* Wl = smem;                 // [128][WSTR]
    float* Xl = smem + 128 * WSTR;    // [16][XSTR]

    const int tid  = threadIdx.x;
    const int row0 = blockIdx.x * 16;

    // Stage W (128x128 f32) into LDS, coalesced float4 loads.
    for (int i = tid; i < 128 * 32; i += 128) {
        int r  = i >> 5;
        int c4 = i & 31;
        v4f v = ((const v4f*)W)[r * 32 + c4];
        *(v4f*)&Wl[r * WSTR + c4 * 4] = v;
    }
    // Stage the 16x128 x tile.
    for (int i = tid; i < 16 * 32; i += 128) {
        int r  = i >> 5;
        int c4 = i & 31;
        v4f v = ((const v4f*)x)[(size_t)(row0 + r) * 32 + c4];
        *(v4f*)&Xl[r * XSTR + c4 * 4] = v;
    }
    __syncthreads();

    const int wave = tid >> 5;
    const int lane = tid & 31;
    const int L    = lane & 15;   // row (A) / col (B) within tile
    const int g    = lane >> 4;   // K sub-slot: lanes 16-31 hold K+2,K+3

    const int col0 = wave * 16;
    const int col1 = (wave + 4) * 16;

    v8f acc0 = {};
    v8f acc1 = {};

    for (int k0 = 0; k0 < IN_F; k0 += 4) {
        const int k = k0 + 2 * g;
        v2f a, b0, b1;
        a.x  = Xl[L * XSTR + k];
        a.y  = Xl[L * XSTR + k + 1];
        b0.x = Wl[k * WSTR + col0 + L];
        b0.y = Wl[(k + 1) * WSTR + col0 + L];
        b1.x = Wl[k * WSTR + col1 + L];
        b1.y = Wl[(k + 1) * WSTR + col1 + L];
        acc0 = __builtin_amdgcn_wmma_f32_16x16x4_f32(false, a, false, b0, (short)0, acc0, false, false);
        acc1 = __builtin_amdgcn_wmma_f32_16x16x4_f32(false, a, false, b1, (short)0, acc1, false, false);
    }

    // D layout: VGPR r, lane -> element (M = r + 8*g, N = L)
    const int mbase = row0 + 8 * g;
    for (int r = 0; r < 8; ++r) {
        h[(size_t)(mbase + r) * HC + col0 + L] = acc0[r];
        h[(size_t)(mbase + r) * HC + col1 + L] = acc1[r];
    }
}

// ---------- kernel 2: a_src[n][h] = <h[n,h,:], att_src[h]>, same for a_dst ----------
// one wave per node, butterfly reduction over the 32 channels of each head
__global__ __launch_bounds__(256) void gat_attn_coeff(const float* __restrict__ h,
                                                      const float* __restrict__ att_src,
                                                      const float* __restrict__ att_dst,
                                                      float* __restrict__ a_src,
                                                      float* __restrict__ a_dst,
                                                      int N) {
    int n    = blockIdx.x * 8 + (threadIdx.x >> 5);
    int lane = threadIdx.x & 31;
    if (n >= N) return;

    float s[HEADS], d[HEADS];
#pragma unroll
    for (int hh = 0; hh < HEADS; ++hh) {
        float v = h[(size_t)n * HC + hh * OUT_F + lane];
        s[hh] = v * att_src[hh * OUT_F + lane];
        d[hh] = v * att_dst[hh * OUT_F + lane];
    }
#pragma unroll
    for (int off = 16; off > 0; off >>= 1) {
#pragma unroll
        for (int hh = 0; hh < HEADS; ++hh) {
            s[hh] += __shfl_xor(s[hh], off, 32);
            d[hh] += __shfl_xor(d[hh], off, 32);
        }
    }
    if (lane == 0) {
#pragma unroll
        for (int hh = 0; hh < HEADS; ++hh) {
            a_src[n * HEADS + hh] = s[hh];
            a_dst[n * HEADS + hh] = d[hh];
        }
    }
}

// ---------- kernel 3: per-edge logits -> segment max (uint-key atomicMax) ----------
__global__ __launch_bounds__(256) void gat_edge_max(const int* __restrict__ ei,
                                                    const float* __restrict__ a_src,
                                                    const float* __restrict__ a_dst,
                                                    unsigned* __restrict__ mkey,
                                                    int E, int N) {
    int e = blockIdx.x * 256 + threadIdx.x;
    if (e >= E + N) return;
    int s, d;
    if (e < E) { s = ei[e]; d = ei[E + e]; } else { s = d = e - E; }
    v4f as = *(const v4f*)&a_src[s * HEADS];
    v4f ad = *(const v4f*)&a_dst[d * HEADS];
#pragma unroll
    for (int hh = 0; hh < HEADS; ++hh) {
        float lr = leaky(as[hh] + ad[hh]);
        atomicMax(&mkey[d * HEADS + hh], fkey(lr));
    }
}

// ---------- kernel 4: denom[dst][h] += exp(e - m[dst][h]) ----------
__global__ __launch_bounds__(256) void gat_edge_denom(const int* __restrict__ ei,
                                                      const float* __restrict__ a_src,
                                                      const float* __restrict__ a_dst,
                                                      const unsigned* __restrict__ mkey,
                                                      float* __restrict__ denom,
                                                      int E, int N) {
    int e = blockIdx.x * 256 + threadIdx.x;
    if (e >= E + N) return;
    int s, d;
    if (e < E) { s = ei[e]; d = ei[E + e]; } else { s = d = e - E; }
    v4f as = *(const v4f*)&a_src[s * HEADS];
    v4f ad = *(const v4f*)&a_dst[d * HEADS];
#pragma unroll
    for (int hh = 0; hh < HEADS; ++hh) {
        float lr = leaky(as[hh] + ad[hh]);
        float m  = funkey(mkey[d * HEADS + hh]);
        atomicAdd(&denom[d * HEADS + hh], __expf(lr - m));
    }
}

// ---------- kernel 5: out[dst] += alpha * h[src]  (one wave per edge) ----------
// lane l owns channels 4l..4l+3, which lie entirely inside head (l >> 3)
__global__ __launch_bounds__(256) void gat_edge_scatter(const int* __restrict__ ei,
                                                        const float* __restrict__ a_src,
                                                        const float* __restrict__ a_dst,
                                                        const unsigned* __restrict__ mkey,
                                                        const float* __restrict__ denom,
                                                        const float* __restrict__ h,
                                                        float* __restrict__ out,
                                                        int E, int N) {
    int e    = blockIdx.x * 8 + (threadIdx.x >> 5);
    int lane = threadIdx.x & 31;
    if (e >= E + N) return;
    int s, d;
    if (e < E) { s = ei[e]; d = ei[E + e]; } else { s = d = e - E; }

    int head  = lane >> 3;
    float lr  = leaky(a_src[s * HEADS + head] + a_dst[d * HEADS + head]);
    float m   = funkey(mkey[d * HEADS + head]);
    float den = denom[d * HEADS + head];
    float alpha = __expf(lr - m) / (den + 1e-16f);

    v4f hv = *(const v4f*)&h[(size_t)s * HC + lane * 4];
    float* o = &out[(size_t)d * HC + lane * 4];
    atomicAdd(o + 0, hv.x * alpha);
    atomicAdd(o + 1, hv.y * alpha);
    atomicAdd(o + 2, hv.z * alpha);
    atomicAdd(o + 3, hv.w * alpha);
}

// ---------- host ----------
extern "C" void kernel_launch(void* const* d_in, const int* in_sizes, int n_in,
                              void* d_out, int out_size, void* d_ws, size_t ws_size,
                              hipStream_t stream) {
    const float* x        = (const float*)d_in[0];
    const int*   ei       = (const int*)d_in[1];   // [2, E] edge_index
    const float* W        = (const float*)d_in[2];
    const float* att_src  = (const float*)d_in[3];
    const float* att_dst  = (const float*)d_in[4];
    const float* bias     = (const float*)d_in[5];
    float*       out      = (float*)d_out;

    const int N = in_sizes[0] / IN_F;   // 100000
    const int E = in_sizes[1] / 2;      // 1600000

    // workspace layout (all fp32 / u32, naturally aligned)
    float*    h     = (float*)d_ws;                       // N*128
    float*    a_src = h + (size_t)N * HC;                 // N*4
    float*    a_dst = a_src + (size_t)N * HEADS;          // N*4
    unsigned* mkey  = (unsigned*)(a_dst + (size_t)N * HEADS); // N*4
    float*    denom = (float*)(mkey + (size_t)N * HEADS);     // N*4

    const int total = N * HC;
    gat_init<<<(total + 255) / 256, 256, 0, stream>>>(out, bias, mkey, denom, total, N * HEADS);

    gat_gemm_wmma<<<N / 16, 128, GEMM_LDS_BYTES, stream>>>(x, W, h);

    gat_attn_coeff<<<(N + 7) / 8, 256, 0, stream>>>(h, att_src, att_dst, a_src, a_dst, N);

    const int ET = E + N;
    gat_edge_max<<<(ET + 255) / 256, 256, 0, stream>>>(ei, a_src, a_dst, mkey, E, N);
    gat_edge_denom<<<(ET + 255) / 256, 256, 0, stream>>>(ei, a_src, a_dst, mkey, denom, E, N);
    gat_edge_scatter<<<(ET + 7) / 8, 256, 0, stream>>>(ei, a_src, a_dst, mkey, denom, h, out, E, N);
}